// RejectionSampler_66443144069404
// MI455X (gfx1250) — compile-verified
//
#include <hip/hip_runtime.h>
#include <hip/hip_bf16.h>
#include <math.h>

#define RS_B 64
#define RS_S 4
#define RS_V 128000
#define RS_T 256                       // threads per row-block (8 waves of 32)
#define RS_ITERS (RS_V / (RS_T * 4))   // 125 float4 tiles per thread
#define RS_NB 4                        // quad buffer: 2 stages in flight + 1 consuming + slack
#define RS_PLACEHOLDER (-1)

// ---- gfx1250 async-to-LDS path (builtin existence proven by round-1 diagnostics) ----
#if defined(__HIP_DEVICE_COMPILE__) && defined(__gfx1250__)
#if __has_builtin(__builtin_amdgcn_global_load_async_to_lds_b128) && \
    __has_builtin(__builtin_amdgcn_s_wait_asynccnt)
#define RS_USE_ASYNC 1
#endif
#endif

#ifdef RS_USE_ASYNC
typedef __attribute__((ext_vector_type(4))) int rs_v4i;
typedef __attribute__((address_space(1))) rs_v4i* rs_gp_v4i;   // global (AS1) int4*
typedef __attribute__((address_space(3))) rs_v4i* rs_lp_v4i;   // LDS (AS3) int4*
__device__ __forceinline__ void rs_async_b128(const void* g, void* l) {
  // per-lane: copies 16 bytes from the global address in g to the LDS address in l, tracked by ASYNCcnt
  __builtin_amdgcn_global_load_async_to_lds_b128(
      (rs_gp_v4i)(unsigned long long)(uintptr_t)g,
      (rs_lp_v4i)(unsigned)(uintptr_t)l,   // the low 32 bits of the generic pointer are the LDS offset
      0, 0);
}
#endif

// ============================ Kernel 1: per-(b,s) row reductions ============================
__global__ __launch_bounds__(RS_T) void rs_row_kernel(
    const float* __restrict__ logits,   // [B,S,V]
    const float* __restrict__ draft,    // [B,S,V]
    const int*   __restrict__ dtok,     // [B,S]
    const float* __restrict__ invq,     // [B,V]
    int*   __restrict__ am_idx,         // [B*S] argmax(logits)
    float* __restrict__ t_at,           // [B*S] softmax prob at draft token
    float* __restrict__ d_at,           // [B*S] draft prob at draft token
    int*   __restrict__ rec_idx)        // [B*S] argmax(max(t-d,0)*invq)
{
  const int row = blockIdx.x;          // b*S + s
  const int b   = row / RS_S;
  const int t   = threadIdx.x;
  const float* L = logits + (size_t)row * RS_V;
  const float* D = draft  + (size_t)row * RS_V;
  const float* Q = invq   + (size_t)b   * RS_V;

  __shared__ float sM[RS_T];
  __shared__ float sS[RS_T];
  __shared__ int   sI[RS_T];
  __shared__ float sBM, sBInv;

  // ---------- pass 1: online softmax + first-occurrence argmax ----------
  // 4 independent accumulator streams (one per float4 lane) to break the
  // serial max->sub->exp->fma dependency chain; merged exactly at the end.
  float am[4] = {-INFINITY, -INFINITY, -INFINITY, -INFINITY};
  float as[4] = {0.0f, 0.0f, 0.0f, 0.0f};
  int   ai[4] = {0, 0, 0, 0};
  for (int k = 0; k < RS_ITERS; ++k) {
    const int i4 = k * RS_T + t;
    __builtin_prefetch((const char*)L + (size_t)i4 * 16 + 8192, 0, 3);  // global_prefetch_b8
    const float4 l4 = ((const float4*)L)[i4];
    const int base = i4 * 4;
    const float vv[4] = {l4.x, l4.y, l4.z, l4.w};
#pragma unroll
    for (int c = 0; c < 4; ++c) {
      const float x  = vv[c];
      const float nm = fmaxf(am[c], x);
      ai[c] = (x > am[c]) ? (base + c) : ai[c];   // strict > keeps first occurrence
      as[c] = as[c] * expf(am[c] - nm) + expf(x - nm);
      am[c] = nm;
    }
  }
  // merge the 4 streams (value ties -> smaller index = first occurrence)
  float m = am[0], sum = as[0];
  int   ami = ai[0];
#pragma unroll
  for (int c = 1; c < 4; ++c) {
    const float M2 = fmaxf(m, am[c]);
    sum = sum * expf(m - M2) + as[c] * expf(am[c] - M2);
    ami = (am[c] > m) ? ai[c] : ((m > am[c]) ? ami : (ami < ai[c] ? ami : ai[c]));
    m = M2;
  }

  sM[t] = m; sS[t] = sum; sI[t] = ami;
  __syncthreads();
  for (int off = RS_T / 2; off > 0; off >>= 1) {
    if (t < off) {
      const float m1 = sM[t], m2 = sM[t + off];
      const float s1 = sS[t], s2 = sS[t + off];
      const int   i1 = sI[t], i2 = sI[t + off];
      const float M  = fmaxf(m1, m2);
      sS[t] = s1 * expf(m1 - M) + s2 * expf(m2 - M);
      sM[t] = M;
      sI[t] = (m2 > m1) ? i2 : ((m1 > m2) ? i1 : (i1 < i2 ? i1 : i2));
    }
    __syncthreads();
  }
  if (t == 0) {
    const float M   = sM[0];
    const float inv = 1.0f / sS[0];
    sBM = M; sBInv = inv;
    const int tok = dtok[row];
    am_idx[row] = sI[0];
    d_at[row]   = D[tok];
    t_at[row]   = expf(L[tok] - M) * inv;
  }
  __syncthreads();
  const float M   = sBM;
  const float inv = sBInv;

  // ---------- pass 2: argmax of max(softmax - draft, 0) * inv_q ----------
  float bv[4] = {-1.0f, -1.0f, -1.0f, -1.0f};   // scores >= 0
  int   bi[4] = {0x7fffffff, 0x7fffffff, 0x7fffffff, 0x7fffffff};

#ifdef RS_USE_ASYNC
  __shared__ __align__(16) float lsL[RS_NB][RS_T * 4];
  __shared__ __align__(16) float lsD[RS_NB][RS_T * 4];
  __shared__ __align__(16) float lsQ[RS_NB][RS_T * 4];
  // prime two stages (pipeline depth 2 -> two HBM-latency windows in flight)
#pragma unroll
  for (int p = 0; p < 2; ++p) {
    const int i4 = p * RS_T + t;
    rs_async_b128(&((const float4*)L)[i4], &lsL[p][t * 4]);
    rs_async_b128(&((const float4*)D)[i4], &lsD[p][t * 4]);
    rs_async_b128(&((const float4*)Q)[i4], &lsQ[p][t * 4]);
  }
  for (int k = 0; k < RS_ITERS; ++k) {
    if (k + 2 < RS_ITERS) {
      const int i4 = (k + 2) * RS_T + t;
      const int nb = (k + 2) % RS_NB;
      rs_async_b128(&((const float4*)L)[i4], &lsL[nb][t * 4]);
      rs_async_b128(&((const float4*)D)[i4], &lsD[nb][t * 4]);
      rs_async_b128(&((const float4*)Q)[i4], &lsQ[nb][t * 4]);
      __builtin_amdgcn_s_wait_asynccnt(6);   // stage k done; k+1,k+2 in flight
    } else if (k + 1 < RS_ITERS) {
      __builtin_amdgcn_s_wait_asynccnt(3);   // stage k done; k+1 in flight
    } else {
      __builtin_amdgcn_s_wait_asynccnt(0);
    }
    const int buf = k % RS_NB;
    // each lane reads back exactly what it staged -> no barrier needed
    const float4 l4 = ((const float4*)&lsL[buf][t * 4])[0];
    const float4 d4 = ((const float4*)&lsD[buf][t * 4])[0];
    const float4 q4 = ((const float4*)&lsQ[buf][t * 4])[0];
#else
  for (int k = 0; k < RS_ITERS; ++k) {
    const int i4g = k * RS_T + t;
    const float4 l4 = ((const float4*)L)[i4g];
    const float4 d4 = ((const float4*)D)[i4g];
    const float4 q4 = ((const float4*)Q)[i4g];
#endif
    const int base = (k * RS_T + t) * 4;
    const float lv[4] = {l4.x, l4.y, l4.z, l4.w};
    const float dv[4] = {d4.x, d4.y, d4.z, d4.w};
    const float qv[4] = {q4.x, q4.y, q4.z, q4.w};
#pragma unroll
    for (int c = 0; c < 4; ++c) {
      const float p  = expf(lv[c] - M) * inv;
      const float sc = fmaxf(p - dv[c], 0.0f) * qv[c];
      if (sc > bv[c]) { bv[c] = sc; bi[c] = base + c; }  // in-order -> first occurrence
    }
  }
  // merge the 4 score streams
  float bestv = bv[0]; int besti = bi[0];
#pragma unroll
  for (int c = 1; c < 4; ++c) {
    if (bv[c] > bestv || (bv[c] == bestv && bi[c] < besti)) { bestv = bv[c]; besti = bi[c]; }
  }

  sM[t] = bestv; sI[t] = besti;
  __syncthreads();
  for (int off = RS_T / 2; off > 0; off >>= 1) {
    if (t < off) {
      const float v2 = sM[t + off]; const int i2 = sI[t + off];
      if (v2 > sM[t] || (v2 == sM[t] && i2 < sI[t])) { sM[t] = v2; sI[t] = i2; }
    }
    __syncthreads();
  }
  if (t == 0) rec_idx[row] = sI[0];
}

// ============================ Kernel 2: per-batch finalize ============================
__global__ void rs_finalize(
    const int*   __restrict__ dtok,    // [B,S]
    const int*   __restrict__ bonus,   // [B]
    const int*   __restrict__ topk,    // [B]
    const float* __restrict__ uni,     // [B,S]
    const int*   __restrict__ am_idx,
    const float* __restrict__ t_at,
    const float* __restrict__ d_at,
    const int*   __restrict__ rec_idx,
    float* __restrict__ out)           // [B*(S+1)] tokens | [B] n_rej | [B] last
{
  const int b = blockIdx.x * blockDim.x + threadIdx.x;
  if (b >= RS_B) return;
  int toks[RS_S + 1];
  if (topk[b] == 1) {  // greedy
    bool rej = false, any = false;
    for (int s = 0; s < RS_S; ++s) {
      const int i  = b * RS_S + s;
      const int am = am_idx[i];
      const bool mm = (dtok[i] != am);
      toks[s] = rej ? RS_PLACEHOLDER : am;
      rej = rej || mm; any = any || mm;
    }
    toks[RS_S] = any ? RS_PLACEHOLDER : bonus[b];
  } else {             // random
    bool rej = false, any = false;
    for (int s = 0; s < RS_S; ++s) {
      const int i = b * RS_S + s;
      const float da = d_at[i], ta = t_at[i];
      const float ratio = (da > 0.0f) ? (ta / da) : 0.0f;
      const bool acc = (da > 0.0f) && (ratio >= uni[i]);
      const int tk = acc ? dtok[i] : rec_idx[i];
      toks[s] = rej ? RS_PLACEHOLDER : tk;
      rej = rej || !acc; any = any || !acc;
    }
    toks[RS_S] = any ? RS_PLACEHOLDER : bonus[b];
  }
  int na = 0;
  for (int j = 0; j <= RS_S; ++j) na += (toks[j] != RS_PLACEHOLDER);
  const int last = toks[na - 1];   // na >= 1 always
  for (int j = 0; j <= RS_S; ++j) out[b * (RS_S + 1) + j] = (float)toks[j];
  out[RS_B * (RS_S + 1) + b]        = (float)((RS_S + 1) - na);
  out[RS_B * (RS_S + 1) + RS_B + b] = (float)last;
}

extern "C" void kernel_launch(void* const* d_in, const int* in_sizes, int n_in,
                              void* d_out, int out_size, void* d_ws, size_t ws_size,
                              hipStream_t stream) {
  const float* logits = (const float*)d_in[0];
  const float* draft  = (const float*)d_in[1];
  const int*   dtok   = (const int*)  d_in[2];
  const int*   bonus  = (const int*)  d_in[3];
  const int*   topk   = (const int*)  d_in[4];
  const float* uni    = (const float*)d_in[5];
  const float* invq   = (const float*)d_in[6];

  int*   am  = (int*)d_ws;
  int*   rec = am + RS_B * RS_S;
  float* ta  = (float*)(rec + RS_B * RS_S);
  float* da  = ta + RS_B * RS_S;

  rs_row_kernel<<<RS_B * RS_S, RS_T, 0, stream>>>(logits, draft, dtok, invq, am, ta, da, rec);
  rs_finalize<<<(RS_B + 63) / 64, 64, 0, stream>>>(dtok, bonus, topk, uni, am, ta, da, rec,
                                                   (float*)d_out);
}